// VolumeRenderer_2817498546813
// MI455X (gfx1250) — compile-verified
//
#include <hip/hip_runtime.h>
#include <math.h>

// Volume rendering (NeRF-style, but with exclusive cumPROD transmittance per the
// reference). B=1, R=65536 rays, S=128 samples. Memory-bound: ~168 MB streamed,
// which fits the MI455X 192 MB L2 -> default (RT) cache policy so timed replays
// hit L2. Four rays per wave32, one at a time; lane l owns samples 4l..4l+3.

#define BIG_F 10000000000.0f
#define S_SAMPLES 128
#define RAYS_PER_WAVE 4

__global__ __launch_bounds__(256)
void volrender_kernel(const float* __restrict__ rf,     // (R, 128, 4) rgb+sigma
                      const float* __restrict__ dirs,   // (R, 3)
                      const float* __restrict__ depth,  // (R, 128) sorted
                      float* __restrict__ out,          // (R, 3)
                      int R)
{
    const int lane          = threadIdx.x & 31;
    const int waveInBlock   = threadIdx.x >> 5;
    const int wavesPerBlock = blockDim.x >> 5;
    const int waveId        = blockIdx.x * wavesPerBlock + waveInBlock;
    const int ray0          = waveId * RAYS_PER_WAVE;

    #pragma unroll
    for (int i = 0; i < RAYS_PER_WAVE; ++i) {
        // Ray index is identical across the wave; make that explicit so the
        // compiler does base-address math in SALU and uses saddr-form loads,
        // and so the dirs load can become a scalar load.
        const int r = __builtin_amdgcn_readfirstlane(ray0 + i);
        if (r >= R) return;

        // ---- CDNA5 prefetch of the next ray's tiles at WGP scope (scope 0:
        // pull into all cache levels). The builtin only emits SCOPE_SE (L2-only,
        // useless once the dataset is L2-resident), so use inline asm. Each lane
        // prefetches a distinct 64B slice -> all cachelines of the next 2KB
        // radiance tile + 512B depth tile are requested in 2 instructions.
        if (i + 1 < RAYS_PER_WAVE && r + 1 < R) {
            const char* prf = (const char*)(rf + (size_t)(r + 1) * (S_SAMPLES * 4));
            const char* pdz = (const char*)(depth + (size_t)(r + 1) * S_SAMPLES);
            asm volatile("global_prefetch_b8 %0, off" :: "v"(prf + lane * 64));
            asm volatile("global_prefetch_b8 %0, off" :: "v"(pdz + lane * 16));
        }

        // ---- depths: one b128 per lane (samples 4l..4l+3), wave covers 512B
        const float4 dz = *(const float4*)(depth + (size_t)r * S_SAMPLES + lane * 4);

        // ---- ray direction norm (wave-uniform; scalar-load friendly)
        const float dxv = dirs[(size_t)r * 3 + 0];
        const float dyv = dirs[(size_t)r * 3 + 1];
        const float dzv = dirs[(size_t)r * 3 + 2];
        const float dn  = sqrtf(dxv * dxv + dyv * dyv + dzv * dzv);

        // ---- gammas: diffs within lane; last diff needs next lane's sample 0
        const float dnext = __shfl_down(dz.x, 1, 32);   // sample 4l+4
        const float g0 = (dz.y - dz.x) * dn;
        const float g1 = (dz.z - dz.y) * dn;
        const float g2 = (dz.w - dz.z) * dn;
        const float g3 = (lane == 31) ? (BIG_F * dn) : (dnext - dz.w) * dn;

        // ---- radiance: 4 x b128 per lane, 64B contiguous per lane;
        // the wave's 4 instructions cover the ray's 2KB block exactly once.
        const float4* rb = (const float4*)(rf + (size_t)r * (S_SAMPLES * 4)) + lane * 4;
        const float4 s0 = rb[0];
        const float4 s1 = rb[1];
        const float4 s2 = rb[2];
        const float4 s3 = rb[3];

        // sigma * gamma
        const float x0 = s0.w * g0;
        const float x1 = s1.w * g1;
        const float x2 = s2.w * g2;
        const float x3 = s3.w * g3;

        // ---- exclusive cumprod over 128 samples:
        // per-lane product -> wave32 inclusive scan -> shift -> in-lane prefix
        const float P = x0 * x1 * x2 * x3;
        float inc = P;
        #pragma unroll
        for (int off = 1; off < 32; off <<= 1) {
            const float t = __shfl_up(inc, off, 32);
            if (lane >= off) inc *= t;
        }
        float exc = __shfl_up(inc, 1, 32);
        if (lane == 0) exc = 1.0f;

        const float e0 = exc;
        const float e1 = e0 * x0;
        const float e2 = e1 * x1;
        const float e3 = e2 * x2;

        // weights = exp(-exclusive_cumprod) * (1 - exp(-x))
        const float w0 = __expf(-e0) * (1.0f - __expf(-x0));
        const float w1 = __expf(-e1) * (1.0f - __expf(-x1));
        const float w2 = __expf(-e2) * (1.0f - __expf(-x2));
        const float w3 = __expf(-e3) * (1.0f - __expf(-x3));

        float accR = w0 * s0.x + w1 * s1.x + w2 * s2.x + w3 * s3.x;
        float accG = w0 * s0.y + w1 * s1.y + w2 * s2.y + w3 * s3.y;
        float accB = w0 * s0.z + w1 * s1.z + w2 * s2.z + w3 * s3.z;

        // ---- wave32 butterfly reduction (all lanes end with the full sum)
        #pragma unroll
        for (int off = 16; off > 0; off >>= 1) {
            accR += __shfl_xor(accR, off, 32);
            accG += __shfl_xor(accG, off, 32);
            accB += __shfl_xor(accB, off, 32);
        }

        // lanes 0..2 store one channel each (single 12B coalesced burst)
        const float myOut = (lane == 0) ? accR : (lane == 1) ? accG : accB;
        if (lane < 3) {
            out[(size_t)r * 3 + lane] = myOut;
        }
    }
}

extern "C" void kernel_launch(void* const* d_in, const int* in_sizes, int n_in,
                              void* d_out, int out_size, void* d_ws, size_t ws_size,
                              hipStream_t stream) {
    const float* rf    = (const float*)d_in[0];  // (B,R,S,4) f32
    const float* dirs  = (const float*)d_in[1];  // (B,R,3)   f32
    const float* depth = (const float*)d_in[2];  // (B,R,S)   f32
    float*       out   = (float*)d_out;          // (B,R,3)   f32

    const int R = in_sizes[1] / 3;               // B*R rays total (B folded in)

    const int threads       = 256;               // 8 waves per block
    const int wavesPerBlock = threads / 32;
    const int raysPerBlock  = wavesPerBlock * RAYS_PER_WAVE;
    const int blocks        = (R + raysPerBlock - 1) / raysPerBlock;

    volrender_kernel<<<blocks, threads, 0, stream>>>(rf, dirs, depth, out, R);
}